// FuzzyMultiLayer_20547123544799
// MI455X (gfx1250) — compile-verified
//
#include <hip/hip_runtime.h>

typedef __attribute__((ext_vector_type(2))) float v2f;
typedef __attribute__((ext_vector_type(8))) float v8f;

#define NPIX (8 * 256 * 256)   // 524288 pixels total (B*H*W)
#define KCLS 10
#define WS_BPACK  0            // [20 combos][32 lanes] B-operand pack
#define WS_TPACK  640          // [5 pairs][32 lanes]  -t seed values
#define WS_LCPACK 800          // [5 pairs][32 lanes]  log-coefficients
#define WS_GAUSST 1024         // gaussT[k][pix] : 10*524288 floats (~21 MB)

// ---------------------------------------------------------------------------
// Setup: invert each lower-triangular L_k, compute t_k = L_k^-1 mu_k and the
// log coefficient; pack B-operands in exact WMMA 4x16 f32 VGPR/lane layout.
// ---------------------------------------------------------------------------
__global__ __launch_bounds__(32) void setup_kernel(const float* __restrict__ scale,
                                                   const float* __restrict__ mean,
                                                   float* __restrict__ ws) {
  __shared__ float Minv[KCLS][8][8];
  __shared__ float tvec[KCLS][8];
  __shared__ float lcv[KCLS];
  int tid = threadIdx.x;
  if (tid < KCLS) {
    const float* L = scale + tid * 64;   // (K,1,C,C) row-major; use lower tri
    float M[8][8];
    for (int i = 0; i < 8; ++i)
      for (int j = 0; j < 8; ++j) M[i][j] = 0.0f;
    for (int j = 0; j < 8; ++j) {
      M[j][j] = 1.0f / L[j * 8 + j];
      for (int i = j + 1; i < 8; ++i) {
        float s = 0.0f;
        for (int m = j; m < i; ++m) s += L[i * 8 + m] * M[m][j];
        M[i][j] = -s / L[i * 8 + i];
      }
    }
    float logdet = 0.0f;
    for (int i = 0; i < 8; ++i) logdet += __logf(fabsf(L[i * 8 + i]));
    for (int i = 0; i < 8; ++i) {
      float t = 0.0f;
      for (int j = 0; j < 8; ++j) t += M[i][j] * mean[tid * 8 + j];
      tvec[tid][i] = t;
      for (int j = 0; j < 8; ++j) Minv[tid][i][j] = M[i][j];
    }
    // -0.5*C*log(2*pi) - logdet,  C=8
    lcv[tid] = -7.3515082656373815f - logdet;
  }
  __syncthreads();
  int l = tid;                 // lane 0..31
  int np = l & 15;             // output column n'
  int half = l >> 4;
  // B-operand pack: combo = p*4 + q*2 + r  (pair p, K-chunk q, vgpr r)
  for (int combo = 0; combo < 20; ++combo) {
    int p = combo >> 2, q = (combo >> 1) & 1, r = combo & 1;
    int kk = r + 2 * half;           // row of 4x16 B tile held by this lane/vgpr
    int ccomp = q * 4 + kk;          // diff component (0..7)
    int cls = 2 * p + ((np >> 3) & 1);
    int ncomp = np & 7;              // z component
    ws[WS_BPACK + combo * 32 + l] = Minv[cls][ncomp][ccomp];  // B[c][n] = M[n][c]
  }
  for (int p = 0; p < 5; ++p) {
    int cls = 2 * p + ((l >> 3) & 1);
    ws[WS_TPACK + p * 32 + l] = tvec[cls][l & 7];
    ws[WS_LCPACK + p * 32 + l] = lcv[cls];
  }
}

// ---------------------------------------------------------------------------
// Phase A: one 16-pixel tile per wave. 10x v_wmma_f32_16x16x4_f32 per tile
// computes z for all 10 classes; transpose-reduce -> per-lane (pixel,class)
// quad; exp; cross-lane normalize; write gauss (ch 0..9), x copy (ch 10..17)
// and the compact gaussT workspace for the gather phase.
// ---------------------------------------------------------------------------
__global__ __launch_bounds__(256) void gauss_kernel(const float* __restrict__ x,
                                                    const float* __restrict__ ws,
                                                    float* __restrict__ out,
                                                    float* __restrict__ gaussT) {
  int l = threadIdx.x & 31;
  int wave = (blockIdx.x << 3) + (threadIdx.x >> 5);
  int tile = wave << 4;                   // first of 16 pixels for this wave
  int half = l >> 4;
  int pix = tile + (l & 15);              // pixel this lane loads (A-operand row)

  const float2* xv = (const float2*)(x + (size_t)pix * 8);
  float2 x0 = xv[half];                   // comps {2h, 2h+1}   -> A K-chunk 0
  float2 x1 = xv[2 + half];               // comps {4+2h, 5+2h} -> A K-chunk 1
  v2f a0; a0[0] = x0.x; a0[1] = x0.y;
  v2f a1; a1[0] = x1.x; a1[1] = x1.y;

  int clsbit = (l >> 3) & 1;
  int pixme  = (l & 7) + (half << 3);     // pixel owned by this lane post-reduce
  int pixout = tile + pixme;

  float prob[5];
  float ss = 0.0f;
#pragma unroll
  for (int p = 0; p < 5; ++p) {
    int base = (p << 2) * 32;
    v2f b0; b0[0] = ws[base + l];      b0[1] = ws[base + 32 + l];
    v2f b1; b1[0] = ws[base + 64 + l]; b1[1] = ws[base + 96 + l];
    float tv = ws[WS_TPACK + (p << 5) + l];
    float lc = ws[WS_LCPACK + (p << 5) + l];
    float ntv = -tv;
    v8f acc = {ntv, ntv, ntv, ntv, ntv, ntv, ntv, ntv};
    acc = __builtin_amdgcn_wmma_f32_16x16x4_f32(false, a0, false, b0,
                                                (short)0, acc, false, false);
    acc = __builtin_amdgcn_wmma_f32_16x16x4_f32(false, a1, false, b1,
                                                (short)0, acc, false, false);
    float s[8];
#pragma unroll
    for (int r2 = 0; r2 < 8; ++r2) { float v = acc[r2]; s[r2] = v * v; }
    // transpose-reduce within each 8-lane group: lane ends with row r = l&7
    bool h4 = (l & 4) != 0;
#pragma unroll
    for (int j = 0; j < 4; ++j) {
      float keep = h4 ? s[j + 4] : s[j];
      float give = h4 ? s[j] : s[j + 4];
      s[j] = keep + __shfl_xor(give, 4);
    }
    bool h2 = (l & 2) != 0;
#pragma unroll
    for (int j = 0; j < 2; ++j) {
      float keep = h2 ? s[j + 2] : s[j];
      float give = h2 ? s[j] : s[j + 2];
      s[j] = keep + __shfl_xor(give, 2);
    }
    bool h1 = (l & 1) != 0;
    {
      float keep = h1 ? s[1] : s[0];
      float give = h1 ? s[0] : s[1];
      s[0] = keep + __shfl_xor(give, 1);
    }
    float pr = __expf(fmaf(s[0], -0.5f, lc));
    prob[p] = pr;
    ss = fmaf(pr, pr, ss);
  }
  // lanes l and l^8 hold the other 5 classes of the same pixel
  ss += __shfl_xor(ss, 8);
  float rn = rsqrtf(fmaxf(ss, 1e-12f));
#pragma unroll
  for (int p = 0; p < 5; ++p) {
    float g = prob[p] * rn;
    int cls = (p << 1) + clsbit;
    out[(size_t)pixout * 98 + cls] = g;                 // channels 0..9
    gaussT[(size_t)cls * NPIX + pixout] = g;            // compact [k][pix]
  }
  // x copy -> channels 10..17 (lanes cooperatively cover all 8 comps)
  float2* ox = (float2*)(out + (size_t)pix * 98 + 10);
  ox[half] = x0;
  ox[2 + half] = x1;
}

// ---------------------------------------------------------------------------
// Phase B: permuted outer-product section (channels 18..97). Destination flat
// index F = destPix*80 + j decodes directly: k=F>>22, c=(F>>19)&7,
// (b*65536+n)=F&0x7FFFF. Writes coalesced; reads are L2-resident gathers.
// ---------------------------------------------------------------------------
__global__ __launch_bounds__(256) void outer_kernel(const float* __restrict__ x,
                                                    const float* __restrict__ gaussT,
                                                    float* __restrict__ out) {
  unsigned gid = blockIdx.x * 256u + threadIdx.x;       // == F
  unsigned k  = gid >> 22;
  unsigned c  = (gid >> 19) & 7u;
  unsigned bn = gid & 0x7FFFFu;                         // b*65536 + n
  float g  = gaussT[(size_t)k * NPIX + bn];
  float xe = x[((size_t)bn << 3) + c];
  unsigned dp = gid / 80u;
  unsigned j  = gid - dp * 80u;
  out[(size_t)dp * 98 + 18 + j] = g * xe;
}

extern "C" void kernel_launch(void* const* d_in, const int* in_sizes, int n_in,
                              void* d_out, int out_size, void* d_ws, size_t ws_size,
                              hipStream_t stream) {
  const float* x     = (const float*)d_in[0];
  const float* scale = (const float*)d_in[1];
  const float* mean  = (const float*)d_in[2];
  float* out = (float*)d_out;
  float* ws  = (float*)d_ws;

  setup_kernel<<<1, 32, 0, stream>>>(scale, mean, ws);
  // 4096 blocks * 8 waves * 16 pixels = 524288 pixels
  gauss_kernel<<<NPIX / 128, 256, 0, stream>>>(x, ws, out, ws + WS_GAUSST);
  // 524288 * 80 = 41,943,040 elements / 256
  outer_kernel<<<(NPIX / 256) * 80, 256, 0, stream>>>(x, ws + WS_GAUSST, out);
}